// BatchAlignmentLoss_53326313947666
// MI455X (gfx1250) — compile-verified
//
#include <hip/hip_runtime.h>
#include <math.h>

#define N_SAMP 8192
#define DIMS   2048
#define PCLS   512
#define KPER   16          // N_SAMP / PCLS instances per class
#define TAU_F  0.5f
#define EPS_F  1e-12f

typedef __attribute__((ext_vector_type(2))) float v2f;
typedef __attribute__((ext_vector_type(8))) float v8f;

// ---------- deterministic block reductions (256 threads) ----------
__device__ __forceinline__ float block_sum256(float v, float* red) {
    int t = threadIdx.x;
    red[t] = v;
    __syncthreads();
#pragma unroll
    for (int s = 128; s > 0; s >>= 1) {
        if (t < s) red[t] += red[t + s];
        __syncthreads();
    }
    float r = red[0];
    __syncthreads();
    return r;
}

__device__ __forceinline__ float block_max256(float v, float* red) {
    int t = threadIdx.x;
    red[t] = v;
    __syncthreads();
#pragma unroll
    for (int s = 128; s > 0; s >>= 1) {
        if (t < s) red[t] = fmaxf(red[t], red[t + s]);
        __syncthreads();
    }
    float r = red[0];
    __syncthreads();
    return r;
}

// ---------- kernel 1: row norms + normalized class centers (one feature pass) ----------
// grid = 3*512 blocks, 256 threads. Block (m,p) holds the full 2048-wide center row
// in registers (8 floats/thread), reading each of its 16 feature rows exactly once.
__global__ void centers_kernel(const float* __restrict__ fv,
                               const float* __restrict__ fa,
                               const float* __restrict__ fr,
                               float* __restrict__ centers,    // [3][512][2048] normalized
                               float* __restrict__ inv_norm)   // [3][8192] per-row 1/max(||x||,eps)
{
    __shared__ float red[256];
    const int bid = blockIdx.x;
    const int m = bid >> 9;          // modality
    const int p = bid & 511;         // class
    const float* f = (m == 0) ? fv : (m == 1) ? fa : fr;
    const int t = threadIdx.x;

    float acc[8];
#pragma unroll
    for (int j = 0; j < 8; ++j) acc[j] = 0.0f;

    for (int k = 0; k < KPER; ++k) {
        const int n = p + k * PCLS;               // labels are arange(N) % 512
        const float* row = f + (size_t)n * DIMS;
        float v[8];
        float ss = 0.0f;
#pragma unroll
        for (int j = 0; j < 8; ++j) {
            v[j] = row[t + 256 * j];
            ss += v[j] * v[j];
        }
        ss = block_sum256(ss, red);
        const float inv = 1.0f / fmaxf(sqrtf(ss), EPS_F);
        if (t == 0) inv_norm[m * N_SAMP + n] = inv;
#pragma unroll
        for (int j = 0; j < 8; ++j) acc[j] += v[j] * inv;
    }

    // center = l2norm(acc * (1/16)) exactly as the reference does it
    float ssc = 0.0f;
#pragma unroll
    for (int j = 0; j < 8; ++j) {
        acc[j] *= (1.0f / 16.0f);
        ssc += acc[j] * acc[j];
    }
    ssc = block_sum256(ssc, red);
    const float invc = 1.0f / fmaxf(sqrtf(ssc), EPS_F);
    float* crow = centers + ((size_t)m * PCLS + p) * DIMS;
#pragma unroll
    for (int j = 0; j < 8; ++j) crow[t + 256 * j] = acc[j] * invc;
}

// ---------- kernel 2: intra dot products (second & last feature pass) ----------
// ||f - c||^2 = 2 - 2 f.c  (both unit vectors); f.c = (x.c) * inv_norm(x)
__global__ void intra_kernel(const float* __restrict__ fv,
                             const float* __restrict__ fa,
                             const float* __restrict__ fr,
                             const int*   __restrict__ label,
                             const float* __restrict__ centers,
                             const float* __restrict__ inv_norm,
                             float* __restrict__ partial)       // [3*8192]
{
    __shared__ float red[256];
    const int bid = blockIdx.x;
    const int m = bid >> 13;          // / 8192
    const int n = bid & (N_SAMP - 1);
    const float* f = (m == 0) ? fv : (m == 1) ? fa : fr;
    const int p = label[n];
    const float* row = f + (size_t)n * DIMS;
    const float* c   = centers + ((size_t)m * PCLS + p) * DIMS;
    const int t = threadIdx.x;

    float d = 0.0f;
#pragma unroll
    for (int j = 0; j < 8; ++j) {
        const int idx = t + 256 * j;
        d += row[idx] * c[idx];
    }
    d = block_sum256(d, red);
    if (t == 0) partial[bid] = d * inv_norm[m * N_SAMP + n];
}

// ---------- kernel 3: center Gram matrices via V_WMMA_F32_16X16X4_F32 ----------
// One wave (32 threads) per 16x16 output tile; K = 2048 -> 512 WMMA steps split
// across two independent accumulator chains. Logits scaled by 1/TAU at store.
__global__ void __launch_bounds__(32)
pair_gemm_kernel(const float* __restrict__ centers,
                 float* __restrict__ G)                         // [3][512][512]
{
    const int bid  = blockIdx.x;      // 3 * 32 * 32
    const int pair = bid >> 10;
    const int tile = bid & 1023;
    const int ti = tile >> 5;
    const int tj = tile & 31;
    // pair 0: cv@ca^T, 1: cv@cr^T, 2: ca@cr^T
    const int ai = (pair == 2) ? 1 : 0;
    const int bi = (pair == 0) ? 1 : 2;
    const float* A = centers + (size_t)ai * PCLS * DIMS;
    const float* B = centers + (size_t)bi * PCLS * DIMS;

    const int lane = threadIdx.x;
    const int r16  = lane & 15;
    const int hi   = lane >> 4;       // 0: K pair {0,1}, 1: K pair {2,3}

    // A tile 16x4: lane holds M=r16, K = 2*hi .. 2*hi+1   (contiguous float2)
    // B tile 4x16 (= rows of C2): lane holds N=r16, K = 2*hi .. 2*hi+1
    const float* arow = A + (size_t)(ti * 16 + r16) * DIMS + 2 * hi;
    const float* brow = B + (size_t)(tj * 16 + r16) * DIMS + 2 * hi;

    v8f acc0 = {};
    v8f acc1 = {};
#pragma unroll 4
    for (int k = 0; k < DIMS; k += 8) {
        v2f a0 = *(const v2f*)(arow + k);
        v2f b0 = *(const v2f*)(brow + k);
        v2f a1 = *(const v2f*)(arow + k + 4);
        v2f b1 = *(const v2f*)(brow + k + 4);
        acc0 = __builtin_amdgcn_wmma_f32_16x16x4_f32(
            /*neg_a=*/false, a0, /*neg_b=*/false, b0,
            /*c_mod=*/(short)0, acc0, /*reuse_a=*/false, /*reuse_b=*/false);
        acc1 = __builtin_amdgcn_wmma_f32_16x16x4_f32(
            false, a1, false, b1, (short)0, acc1, false, false);
    }

    float* g = G + (size_t)pair * PCLS * PCLS;
#pragma unroll
    for (int r = 0; r < 8; ++r) {
        const int row = ti * 16 + r + 8 * hi;   // C/D layout: VGPR r, lanes 16-31 = M+8
        const int col = tj * 16 + r16;
        g[(size_t)row * PCLS + col] = (acc0[r] + acc1[r]) * (1.0f / TAU_F);
    }
}

// ---------- kernel 4: per-row log-softmax diagonal term ----------
__global__ void softmax_diag_kernel(const float* __restrict__ G,
                                    float* __restrict__ pair_partial)  // [3*512]
{
    __shared__ float red[256];
    const int bid  = blockIdx.x;      // 3*512
    const int pair = bid >> 9;
    const int i    = bid & 511;
    const float* row = G + ((size_t)pair * PCLS + i) * PCLS;
    const int t = threadIdx.x;

    const float x0 = row[t];
    const float x1 = row[t + 256];
    const float mx = block_max256(fmaxf(x0, x1), red);
    const float se = block_sum256(__expf(x0 - mx) + __expf(x1 - mx), red);
    if (t == 0) {
        const float diag = row[i];
        // -logp[i][i] = max + log(sum_exp) - logits[i][i]
        pair_partial[bid] = mx + logf(se) - diag;
    }
}

// ---------- kernel 5: final combine ----------
__global__ void final_kernel(const float* __restrict__ intra_partial,  // [3*8192] f.c dots
                             const float* __restrict__ pair_partial,   // [3*512] -logp diag
                             float* __restrict__ out)
{
    __shared__ float red[256];
    const int t = threadIdx.x;
    float s = 0.0f;
    for (int idx = t; idx < 3 * N_SAMP; idx += 256) s += intra_partial[idx];
    s = block_sum256(s, red);
    float q = 0.0f;
    for (int idx = t; idx < 3 * PCLS; idx += 256) q += pair_partial[idx];
    q = block_sum256(q, red);
    if (t == 0) {
        const float loss_intra = 6.0f - (2.0f / (float)N_SAMP) * s;
        const float loss_inter = q * (1.0f / (float)PCLS);
        out[0] = loss_intra + loss_inter;
    }
}

extern "C" void kernel_launch(void* const* d_in, const int* in_sizes, int n_in,
                              void* d_out, int out_size, void* d_ws, size_t ws_size,
                              hipStream_t stream) {
    const float* fv    = (const float*)d_in[0];
    const float* fa    = (const float*)d_in[1];
    const float* fr    = (const float*)d_in[2];
    const int*   label = (const int*)d_in[3];

    float* ws        = (float*)d_ws;
    float* centers   = ws;                                   // 3*512*2048
    float* inv_norm  = centers  + (size_t)3 * PCLS * DIMS;   // 3*8192
    float* G         = inv_norm + (size_t)3 * N_SAMP;        // 3*512*512
    float* intra_p   = G        + (size_t)3 * PCLS * PCLS;   // 3*8192
    float* pair_p    = intra_p  + (size_t)3 * N_SAMP;        // 3*512
    // total: ~15.93 MB of d_ws

    centers_kernel<<<3 * PCLS, 256, 0, stream>>>(fv, fa, fr, centers, inv_norm);
    intra_kernel<<<3 * N_SAMP, 256, 0, stream>>>(fv, fa, fr, label, centers, inv_norm, intra_p);
    pair_gemm_kernel<<<3 * 32 * 32, 32, 0, stream>>>(centers, G);
    softmax_diag_kernel<<<3 * PCLS, 256, 0, stream>>>(G, pair_p);
    final_kernel<<<1, 256, 0, stream>>>(intra_p, pair_p, (float*)d_out);
}